// TIRA_32710470926860
// MI455X (gfx1250) — compile-verified
//
#include <hip/hip_runtime.h>
#include <hip/hip_bf16.h>

typedef __attribute__((ext_vector_type(2))) float v2f;
typedef __attribute__((ext_vector_type(8))) float v8f;

// ---------------------------------------------------------------------------
// Native f32 atomic add (global_atomic_add_f32, no CAS loop)
// ---------------------------------------------------------------------------
__device__ __forceinline__ void atomAddF(float* p, float v) {
    unsafeAtomicAdd(p, v);
}

// ---------------------------------------------------------------------------
// CDNA5 async global->LDS copies (ASYNCcnt-tracked, bypass VGPRs).
// LDS operand: low 32 bits of a generic __shared__ pointer are the
// wave-relative LDS byte offset (ISA 10.2 aperture mapping).
// ---------------------------------------------------------------------------
__device__ __forceinline__ unsigned lds_off(const void* p) {
    return (unsigned)(unsigned long long)p;
}

__device__ __forceinline__ void async_b128(unsigned ldsOff, const float* g) {
    asm volatile("global_load_async_to_lds_b128 %0, %1, off"
                 :: "v"(ldsOff), "v"(g) : "memory");
}

__device__ __forceinline__ void async_b32(unsigned ldsOff, const float* g) {
    asm volatile("global_load_async_to_lds_b32 %0, %1, off"
                 :: "v"(ldsOff), "v"(g) : "memory");
}

template <int N>
__device__ __forceinline__ void wait_async() {
#if __has_builtin(__builtin_amdgcn_s_wait_asynccnt)
    __builtin_amdgcn_s_wait_asynccnt(N);
#else
    asm volatile("s_wait_asynccnt %0" :: "i"(N) : "memory");
#endif
}

// ---------------------------------------------------------------------------
// Degree / norm kernels:  deg[i] = 1 (self loop) + #edges with src==i
//                         norm[i] = deg[i]^-1/2
// ---------------------------------------------------------------------------
__global__ void k_init_deg(float* __restrict__ deg, int n) {
    int i = blockIdx.x * blockDim.x + threadIdx.x;
    if (i < n) deg[i] = 1.0f;
}

__global__ void k_count_src(const int* __restrict__ src, float* __restrict__ deg, int e) {
    int i = blockIdx.x * blockDim.x + threadIdx.x;
    if (i < e) atomAddF(&deg[src[i]], 1.0f);
}

__global__ void k_norm(float* __restrict__ deg, int n) {
    int i = blockIdx.x * blockDim.x + threadIdx.x;
    if (i < n) deg[i] = rsqrtf(deg[i]);
}

// ---------------------------------------------------------------------------
// Aggregation init (pure write, runs BEFORE the atomic scatter):
//   out[i][c] = bias[c] + h[i][c] * norm[i]^2        (bias + self-loop term)
// ---------------------------------------------------------------------------
__global__ void k_selfinit(const float* __restrict__ h, const float* __restrict__ norm,
                           const float* __restrict__ bias, float* __restrict__ out,
                           int N, int H) {
    long long idx = (long long)blockIdx.x * blockDim.x + threadIdx.x;
    long long total = (long long)N * H;
    if (idx >= total) return;
    int i = (int)(idx / H);
    int c = (H & (H - 1)) == 0 ? (int)(idx & (H - 1)) : (int)(idx % H);
    float w = norm[i];
    out[idx] = bias[c] + h[idx] * (w * w);
}

// ---------------------------------------------------------------------------
// Edge aggregation: out[dst] += h[src] * norm[src]*norm[dst]
// one wave per edge; lane moves float4 chunks of the H-wide row (H % 4 == 0)
// ---------------------------------------------------------------------------
__global__ __launch_bounds__(256) void k_agg(const float* __restrict__ h,
                                             const int* __restrict__ src,
                                             const int* __restrict__ dst,
                                             const float* __restrict__ norm,
                                             float* __restrict__ out,
                                             int E, int H) {
    int e = blockIdx.x * (blockDim.x >> 5) + (threadIdx.x >> 5);
    if (e >= E) return;
    int lane = threadIdx.x & 31;
    int s = src[e], d = dst[e];
    float w = norm[s] * norm[d];
    const float* hs = h + (size_t)s * H;
    float* od = out + (size_t)d * H;
    for (int j = lane * 4; j < H; j += 128) {
        float4 v = *(const float4*)(hs + j);
        atomAddF(od + j + 0, v.x * w);
        atomAddF(od + j + 1, v.y * w);
        atomAddF(od + j + 2, v.z * w);
        atomAddF(od + j + 3, v.w * w);
    }
}

// h[i] = relu(h2[i]) + relu(h1pre[i])   (h1 stored pre-relu)
__global__ void k_relu_add2(const float* __restrict__ h2, const float* __restrict__ h1p,
                            float* __restrict__ out, long long n) {
    long long idx = (long long)blockIdx.x * blockDim.x + threadIdx.x;
    if (idx < n) out[idx] = fmaxf(h2[idx], 0.0f) + fmaxf(h1p[idx], 0.0f);
}

// ---------------------------------------------------------------------------
// Async double-buffered fp32 WMMA GEMM:
//   C[M,N] = act(reluA(A)[M,K] @ B[K,N] + colBias + resMat),  row-major.
// Requires N%(16*NT)==0, K%32==0 (all shapes here qualify); M ragged per-wave.
//
// Block = 8 waves, block tile 128(M) x 16*NT(N), K chunked by KC=32.
// Chunk c+1 is fetched with global_load_async_to_lds_b* while chunk c is
// consumed by WMMA; s_wait_asynccnt <= ASY retires exactly the current chunk
// (async copies complete in order; every wave issues ASY copies per chunk).
//
//  - As natural [128][RSA=KC+4]: A-fragment ds_load_b64 covers all 64 banks.
//  - Bs pair-interleaved via per-lane b32 async scatter:
//      B[kc+2p+q][n0+c]  ->  Bs[p*RSB + 2c + q],  RSB % 64 == 32 for NT=4 (64 for NT=1).
//    A B fragment is then ONE contiguous ds_load_b64 (no packing moves);
//    for NT=4, lane halves (pair rows p, p+1) land on disjoint bank ranges.
//
// RELUA is a compile-time flag so non-relu GEMMs carry zero A-side overhead.
//
// Fragment layouts (ISA 7.12.2):
//   A 16x4 : lane l holds A[l&15][klo + v],  klo = 2*(l>=16), v = vgpr 0..1
//   B 4x16 : lane l holds B[klo + v][l&15]
//   C 16x16: vgpr r, lane l -> C[r + 8*(l>=16)][l&15]
// ---------------------------------------------------------------------------
template <int NT, bool RELUA>
__global__ __launch_bounds__(256) void k_gemm_async(const float* __restrict__ A,
                                                    const float* __restrict__ B,
                                                    float* __restrict__ C,
                                                    int M, int N, int K,
                                                    const float* __restrict__ colBias,
                                                    const float* __restrict__ resMat,
                                                    int doRelu) {
    constexpr int KC  = 32;
    constexpr int RSA = KC + 4;                 // 36: rc*36 mod 64 -> 16 distinct mult-of-4
    constexpr int RSB = 16 * NT * 2 + 32;       // 160 (NT=4, %64==32) / 64 (NT=1, %64==0)
    constexpr int ASY = (NT == 4) ? 12 : 6;     // async instrs per wave per chunk
    __shared__ float As[2][128 * RSA];
    __shared__ float Bs[2][(KC / 2) * RSB];

    const int tid    = threadIdx.x;
    const int wave   = tid >> 5;
    const int lane   = tid & 31;
    const int mblock = blockIdx.y * 128;
    const int m0     = mblock + wave * 16;
    const int n0     = blockIdx.x * (16 * NT);
    const int rc     = lane & 15;
    const int khi    = (lane >> 4) << 1;
    const bool active = (m0 < M);

    // issue one chunk's async copies into buffer `buf` (wave-uniform: ASY per wave)
    auto issue = [&](int buf, int kc) {
        // A tile: 128 x KC floats, b128 chunks -> 4 instrs/wave
        for (int i = tid; i < 128 * (KC / 4); i += 256) {
            int r  = i >> 3;                 // KC/4 == 8 chunks per row
            int c4 = (i & 7) * 4;
            int gr = mblock + r; if (gr >= M) gr = M - 1;
            async_b128(lds_off(&As[buf][r * RSA + c4]),
                       A + (size_t)gr * K + kc + c4);
        }
        // B tile: KC x 16*NT elements, per-lane b32 scatter into pair-interleave
        // NT=4: 2048 -> 8 instrs/wave ; NT=1: 512 -> 2 instrs/wave
        for (int i = tid; i < KC * 16 * NT; i += 256) {
            int r = i / (16 * NT);
            int c = i - r * (16 * NT);
            async_b32(lds_off(&Bs[buf][(r >> 1) * RSB + 2 * c + (r & 1)]),
                      B + (size_t)(kc + r) * N + n0 + c);
        }
    };

    v8f acc[NT] = {};
    const int nch = K / KC;

    issue(0, 0);
    for (int c = 0; c < nch; ++c) {
        const int cur = c & 1;
        if (c + 1 < nch) {
            issue(cur ^ 1, (c + 1) * KC);    // prefetch next chunk
            wait_async<ASY>();               // retire current chunk only
        } else {
            wait_async<0>();
        }
        __syncthreads();

        if (active) {
            const float* ap = &As[cur][(wave * 16 + rc) * RSA + khi];
#pragma unroll
            for (int k = 0; k < KC; k += 4) {
                v2f av = *(const v2f*)(ap + k);
                if constexpr (RELUA) {
                    av.x = fmaxf(av.x, 0.0f);
                    av.y = fmaxf(av.y, 0.0f);
                }
                const float* bp = &Bs[cur][((k + khi) >> 1) * RSB + rc * 2];
#pragma unroll
                for (int t = 0; t < NT; ++t) {
                    v2f bv = *(const v2f*)(bp + t * 32);
                    acc[t] = __builtin_amdgcn_wmma_f32_16x16x4_f32(
                        false, av, false, bv, (short)0, acc[t], false, false);
                }
            }
        }
        __syncthreads();                     // cur fully read before it is refilled
    }

    if (!active) return;
    const int mrow = m0 + ((lane >> 4) << 3);
#pragma unroll
    for (int t = 0; t < NT; ++t) {
        const int col = n0 + t * 16 + rc;
        const float cb = colBias ? colBias[col] : 0.0f;
#pragma unroll
        for (int r = 0; r < 8; ++r) {
            size_t idx = (size_t)(mrow + r) * N + col;
            float v = acc[t][r] + cb;
            if (resMat) v += resMat[idx];
            if (doRelu) v = fmaxf(v, 0.0f);
            C[idx] = v;
        }
    }
}

// ---------------------------------------------------------------------------
static inline void launch_gemm(const float* A, const float* B, float* C,
                               int M, int N, int K,
                               const float* colBias, const float* resMat,
                               int doRelu, int reluA, hipStream_t stream) {
    dim3 blk(256);
    int gy = (M + 127) / 128;
    if ((N % 64) == 0) {
        dim3 grd(N / 64, gy);
        if (reluA)
            k_gemm_async<4, true><<<grd, blk, 0, stream>>>(A, B, C, M, N, K,
                                                           colBias, resMat, doRelu);
        else
            k_gemm_async<4, false><<<grd, blk, 0, stream>>>(A, B, C, M, N, K,
                                                            colBias, resMat, doRelu);
    } else {
        dim3 grd(N / 16, gy);
        k_gemm_async<1, false><<<grd, blk, 0, stream>>>(A, B, C, M, N, K,
                                                        colBias, resMat, doRelu);
    }
}

static inline int cdiv(long long a, int b) { return (int)((a + b - 1) / b); }

extern "C" void kernel_launch(void* const* d_in, const int* in_sizes, int n_in,
                              void* d_out, int out_size, void* d_ws, size_t ws_size,
                              hipStream_t stream) {
    // --- shapes from input sizes (setup_inputs order) ---
    const int H  = in_sizes[3];            // 128
    const int D  = in_sizes[2] / H;        // 256
    const int N  = in_sizes[0] / D;        // 50000
    const int E  = in_sizes[1] / 2;        // 1.6M
    const int FC = in_sizes[11];           // 1024
    const int NC = in_sizes[13];           // 16

    const float* x      = (const float*)d_in[0];
    const int*   ei     = (const int*)d_in[1];
    const int*   src    = ei;
    const int*   dst    = ei + E;
    const float* gcn0_W = (const float*)d_in[2];
    const float* gcn0_b = (const float*)d_in[3];
    const float* gcn1_W = (const float*)d_in[4];
    const float* gcn1_b = (const float*)d_in[5];
    const float* gcn2_W = (const float*)d_in[6];
    const float* gcn2_b = (const float*)d_in[7];
    const float* w1     = (const float*)d_in[8];
    const float* b1     = (const float*)d_in[9];
    const float* fc0_W  = (const float*)d_in[10];
    const float* fc0_b  = (const float*)d_in[11];
    const float* fc1_W  = (const float*)d_in[12];
    const float* fc1_b  = (const float*)d_in[13];
    float* out = (float*)d_out;

    // --- workspace layout (floats) ---
    float* ws = (float*)d_ws;
    size_t off = 0;
    float* normv = ws + off; off += ((size_t)N + 255) & ~(size_t)255;
    float* bufA  = ws + off; off += (size_t)N * H;   // temps / h
    float* bufB  = ws + off; off += (size_t)N * H;   // h1 (pre-relu)
    float* bufC  = ws + off; off += (size_t)N * H;   // h11 / h2
    float* bufFC = ws + off; off += (size_t)N * FC;  // fc0 activations

    const long long NH = (long long)N * H;
    const int T = 256;

    // 1) norm = (1 + out-degree)^-1/2
    k_init_deg<<<cdiv(N, T), T, 0, stream>>>(normv, N);
    k_count_src<<<cdiv(E, T), T, 0, stream>>>(src, normv, E);
    k_norm<<<cdiv(N, T), T, 0, stream>>>(normv, N);

    // 2) gcn0: bufA = x @ W0 ; bufB = bias + self + scatter(bufA)   (pre-relu h1)
    launch_gemm(x, gcn0_W, bufA, N, H, D, nullptr, nullptr, 0, 0, stream);
    k_selfinit<<<cdiv(NH, T), T, 0, stream>>>(bufA, normv, gcn0_b, bufB, N, H);
    k_agg<<<cdiv(E, 8), T, 0, stream>>>(bufA, src, dst, normv, bufB, E, H);

    // 3) h11 = relu(relu(h1pre) @ w1 + b1)   (reluA folds gcn0's activation)
    launch_gemm(bufB, w1, bufC, N, H, H, nullptr, b1, 1, 1, stream);       // bufC = h11

    // 4) gcn1 (no activation): bufA = h11 @ W1 ; bufC = bias + self + scatter
    launch_gemm(bufC, gcn1_W, bufA, N, H, H, nullptr, nullptr, 0, 0, stream);
    k_selfinit<<<cdiv(NH, T), T, 0, stream>>>(bufA, normv, gcn1_b, bufC, N, H);
    k_agg<<<cdiv(E, 8), T, 0, stream>>>(bufA, src, dst, normv, bufC, E, H); // bufC = h2pre

    // 5) gcn2: bufA = h2pre @ W2 ; bufC = bias + self + scatter
    launch_gemm(bufC, gcn2_W, bufA, N, H, H, nullptr, nullptr, 0, 0, stream);
    k_selfinit<<<cdiv(NH, T), T, 0, stream>>>(bufA, normv, gcn2_b, bufC, N, H);
    k_agg<<<cdiv(E, 8), T, 0, stream>>>(bufA, src, dst, normv, bufC, E, H);

    // 6) h = relu(gcn2_out) + relu(h1pre)
    k_relu_add2<<<cdiv(NH, T), T, 0, stream>>>(bufC, bufB, bufA, NH);      // bufA = h

    // 7) fc head
    launch_gemm(bufA, fc0_W, bufFC, N, FC, H, fc0_b, nullptr, 1, 0, stream);
    launch_gemm(bufFC, fc1_W, out, N, NC, FC, fc1_b, nullptr, 0, 0, stream);

    (void)n_in; (void)out_size; (void)ws_size;
}